// NodeLayer_7550552506736
// MI455X (gfx1250) — compile-verified
//
#include <hip/hip_runtime.h>
#include <math.h>

#define EMB 128
#define BN_EPS 1e-5f
#define BSTRIDE 144  // LDS row stride in floats: 2-row offset = 32 banks -> no half-wave conflicts

typedef __attribute__((ext_vector_type(2))) float v2f;
typedef __attribute__((ext_vector_type(8))) float v8f;

// Monotone order-preserving float<->uint encoding so atomicMax(u32) == float max.
__device__ __forceinline__ unsigned enc_f32(float f) {
  unsigned u = __float_as_uint(f);
  return (u & 0x80000000u) ? ~u : (u | 0x80000000u);
}
__device__ __forceinline__ float dec_f32(unsigned u) {
  u = (u & 0x80000000u) ? (u & 0x7fffffffu) : ~u;
  return __uint_as_float(u);
}

__global__ void zero_kernel(unsigned* __restrict__ p, size_t n) {
  size_t i = (size_t)blockIdx.x * blockDim.x + threadIdx.x;
  if (i < n) p[i] = 0u;
}

// One wave32 per edge: lane loads float4 of src row + dst row (32*16B = full 512B row).
__global__ void edge_score_kernel(const float* __restrict__ emb,
                                  const int* __restrict__ src,
                                  const int* __restrict__ dst,
                                  float* __restrict__ scores,
                                  unsigned* __restrict__ segmax,
                                  int E) {
  int wave = (int)((blockIdx.x * blockDim.x + threadIdx.x) >> 5);
  int lane = threadIdx.x & 31;
  if (wave >= E) return;
  int s = src[wave], d = dst[wave];
  const float4* ps = (const float4*)(emb + (size_t)s * EMB);
  const float4* pd = (const float4*)(emb + (size_t)d * EMB);
  float4 a = ps[lane];
  float4 b = pd[lane];
  float v = a.x * b.x + a.y * b.y + a.z * b.z + a.w * b.w;
#pragma unroll
  for (int off = 16; off > 0; off >>= 1) v += __shfl_xor(v, off, 32);
  if (lane == 0) {
    scores[wave] = v;
    atomicMax(segmax + d, enc_f32(v));
  }
}

__global__ void edge_exp_kernel(const int* __restrict__ dst,
                                float* __restrict__ sc,          // in: score, out: exp
                                const unsigned* __restrict__ segmax,
                                float* __restrict__ denom, int E) {
  int e = blockIdx.x * blockDim.x + threadIdx.x;
  if (e >= E) return;
  int d = dst[e];
  float ex = expf(sc[e] - dec_f32(segmax[d]));
  sc[e] = ex;
  atomicAdd(denom + d, ex);
}

// Wave per edge: neigh[dst] += emb[src] * (ex/denom[dst]); 4 f32 L2 atomics per lane.
__global__ void edge_aggr_kernel(const float* __restrict__ emb,
                                 const int* __restrict__ src,
                                 const int* __restrict__ dst,
                                 const float* __restrict__ ex,
                                 const float* __restrict__ denom,
                                 float* __restrict__ neigh, int E) {
  int wave = (int)((blockIdx.x * blockDim.x + threadIdx.x) >> 5);
  int lane = threadIdx.x & 31;
  if (wave >= E) return;
  int s = src[wave], d = dst[wave];
  float norm = ex[wave] / denom[d];
  const float4* ps = (const float4*)(emb + (size_t)s * EMB);
  float4 a = ps[lane];
  float* out = neigh + (size_t)d * EMB + lane * 4;
  atomicAdd(out + 0, a.x * norm);
  atomicAdd(out + 1, a.y * norm);
  atomicAdd(out + 2, a.z * norm);
  atomicAdd(out + 3, a.w * norm);
}

// H = neigh @ W via V_WMMA_F32_16X16X4_F32. Each wave owns a 16(M)x128(N) stripe:
// 8 accumulator tiles, 32 K-steps of K=4. W staged in LDS (stride-144 padded).
// Fused epilogue: per-column sum / sum-of-squares partials for batchnorm.
__global__ void __launch_bounds__(256)
gemm_bn_kernel(const float* __restrict__ A,     // neigh [N,128]
               const float* __restrict__ W,     // [128,128]
               float* __restrict__ H,           // d_out [N,128]
               float* __restrict__ colsum,
               float* __restrict__ colsumsq,
               int nTilesM) {
  __shared__ float Bs[128 * BSTRIDE];
  int tid = threadIdx.x;
  for (int i = tid; i < 128 * 128; i += 256) {
    Bs[(i >> 7) * BSTRIDE + (i & 127)] = W[i];
  }
  __syncthreads();

  int mTile = blockIdx.x * 8 + (tid >> 5);
  if (mTile >= nTilesM) return;  // uniform per wave -> EXEC stays all-ones for WMMA
  int lane = tid & 31;
  int m0 = mTile * 16;
  int rowA = m0 + (lane & 15);        // A 16x4 frag: lanes hold M = lane%16
  int koff = (lane >> 4) * 2;         // lanes 0-15: K=0,1 ; lanes 16-31: K=2,3
  int ncol = lane & 15;               // B/C/D column within tile

  v8f acc[8] = {};
  const float* Arow = A + (size_t)rowA * EMB;

  for (int kk = 0; kk < 32; ++kk) {
    int k0 = kk * 4 + koff;
    v2f a;
    a.x = Arow[k0];
    a.y = Arow[k0 + 1];
#pragma unroll
    for (int t = 0; t < 8; ++t) {
      v2f b;
      b.x = Bs[k0 * BSTRIDE + t * 16 + ncol];
      b.y = Bs[(k0 + 1) * BSTRIDE + t * 16 + ncol];
      acc[t] = __builtin_amdgcn_wmma_f32_16x16x4_f32(
          /*neg_a=*/false, a, /*neg_b=*/false, b,
          /*c_mod=*/(short)0, acc[t], /*reuse_a=*/false, /*reuse_b=*/false);
    }
  }

  // C/D layout: VGPR r -> M = m0 + r (+8 for lanes 16-31), N = tile*16 + lane%16
  int rBase = (lane < 16) ? 0 : 8;
#pragma unroll
  for (int t = 0; t < 8; ++t) {
    float s = 0.f, sq = 0.f;
#pragma unroll
    for (int r = 0; r < 8; ++r) {
      float v = acc[t][r];
      H[(size_t)(m0 + rBase + r) * EMB + t * 16 + ncol] = v;
      s += v;
      sq += v * v;
    }
    // lane L and L+16 hold the same column -> fold, then 16 lanes do the atomics
    s += __shfl_xor(s, 16, 32);
    sq += __shfl_xor(sq, 16, 32);
    if (lane < 16) {
      atomicAdd(colsum + t * 16 + ncol, s);
      atomicAdd(colsumsq + t * 16 + ncol, sq);
    }
  }
}

__global__ void bn_tanh_kernel(float* __restrict__ H,
                               const float* __restrict__ colsum,
                               const float* __restrict__ colsumsq,
                               const float* __restrict__ gamma,
                               const float* __restrict__ beta,
                               size_t total, float invN) {
  size_t i = (size_t)blockIdx.x * blockDim.x + threadIdx.x;
  if (i >= total) return;
  int c = (int)(i & (EMB - 1));
  float mu = colsum[c] * invN;
  float var = colsumsq[c] * invN - mu * mu;  // biased variance, matches reference
  float x = H[i];
  float y = (x - mu) * rsqrtf(var + BN_EPS) * gamma[c] + beta[c];
  H[i] = tanhf(y);
}

extern "C" void kernel_launch(void* const* d_in, const int* in_sizes, int n_in,
                              void* d_out, int out_size, void* d_ws, size_t ws_size,
                              hipStream_t stream) {
  const float* emb   = (const float*)d_in[0];
  const float* W     = (const float*)d_in[1];
  const float* gamma = (const float*)d_in[2];
  const float* beta  = (const float*)d_in[3];
  const int* src     = (const int*)d_in[4];
  const int* dst     = (const int*)d_in[5];
  int N = in_sizes[0] / EMB;     // 50000
  int E = in_sizes[4];           // 1.6M

  // Workspace layout (floats): [ S:E | segmax:N | denom:N | neigh:N*128 | colsum:128 | colsumsq:128 ]
  float*    S        = (float*)d_ws;
  unsigned* segmax   = (unsigned*)(S + E);
  float*    denom    = (float*)(segmax + N);
  float*    neigh    = denom + N;
  float*    colsum   = neigh + (size_t)N * EMB;
  float*    colsumsq = colsum + EMB;

  // Zero everything that is reduced into (segmax..colsumsq are contiguous).
  size_t zeroN = (size_t)N * 2 + (size_t)N * EMB + 2 * EMB;
  zero_kernel<<<(unsigned)((zeroN + 255) / 256), 256, 0, stream>>>(segmax, zeroN);

  edge_score_kernel<<<(E + 7) / 8, 256, 0, stream>>>(emb, src, dst, S, segmax, E);
  edge_exp_kernel<<<(E + 255) / 256, 256, 0, stream>>>(dst, S, segmax, denom, E);
  edge_aggr_kernel<<<(E + 7) / 8, 256, 0, stream>>>(emb, src, dst, S, denom, neigh, E);

  int nTilesM = N / 16;  // 3125, exact for this workload
  gemm_bn_kernel<<<(nTilesM + 7) / 8, 256, 0, stream>>>(neigh, W, (float*)d_out,
                                                        colsum, colsumsq, nTilesM);

  size_t total = (size_t)N * EMB;
  bn_tanh_kernel<<<(unsigned)((total + 255) / 256), 256, 0, stream>>>(
      (float*)d_out, colsum, colsumsq, gamma, beta, total, 1.0f / (float)N);
}